// IHaarDWT_28784870817852
// MI455X (gfx1250) — compile-verified
//
#include <hip/hip_runtime.h>

// Inverse 5-level Haar DWT, fully fused into a single bandwidth-bound pass.
//
// out[j] = (((((a ± d5)*k ± d4)*k ± d3)*k ± d2)*k ± d1)*k
//   a  = x[j>>5]            (approx, level 5)
//   d5 = x[L/32 + (j>>5)]   sign = (-1)^bit4(j)
//   d4 = x[L/16 + (j>>4)]   sign = (-1)^bit3(j)
//   d3 = x[L/8  + (j>>3)]   sign = (-1)^bit2(j)
//   d2 = x[L/4  + (j>>2)]   sign = (-1)^bit1(j)
//   d1 = x[L/2  + (j>>1)]   sign = (-1)^bit0(j)
// Operation order (add, then mul by 1/sqrt2) matches the reference per level.

#define INV_SQRT2_F 0.70710678118654752440f

__global__ __launch_bounds__(256) void ihaar5_fused_kernel(
    const float* __restrict__ x, float* __restrict__ out, unsigned int L) {
  __shared__ __align__(16) float lds[1024];

  const unsigned int t = threadIdx.x;   // 0..255
  const unsigned int g = blockIdx.x;    // workgroup -> 1024 outputs at g*1024

  // ---- Stage this WG's entire input footprint (4 KB, six contiguous
  // segments) into LDS with CDNA5 async global->LDS copies (ASYNCcnt). ----
  // LDS float layout: a[0,32) d5[32,64) d4[64,128) d3[128,256) d2[256,512) d1[512,1024)
  //
  // Branchless segment math: thread ranges are power-of-two aligned.
  //   u = t|8, h = msb(u) in [3,7]
  //   t in [2^h, 2^(h+1)) -> detail level (8-h): base L>>(8-h), stride 32<<(h-3)
  //   t in [0,8)          -> approx: base 0, same stride/offset as h=3
  const unsigned int u = t | 8u;
  const unsigned int h = 31u - (unsigned int)__clz((int)u);       // 3..7
  const unsigned int strideBlocks = 32u << (h - 3u);              // 32..512
  const unsigned int local4 = (u - (1u << h)) * 4u;               // float offset
  const unsigned int base = (t < 8u) ? 0u : (L >> (8u - h));
  const unsigned int goff = base + g * strideBlocks + local4;     // < 2^25, fits u32

  const float* gsrc = x + goff;                       // 16B aligned
  unsigned int lds_byte_off =
      (unsigned int)(size_t)(&lds[t * 4u]);           // low 32 bits = LDS offset

  // 16 bytes per lane: one async b128 per wave moves 512B; 8 waves fill 4KB.
  asm volatile("global_load_async_to_lds_b128 %0, %1, off"
               :
               : "v"(lds_byte_off), "v"(gsrc)
               : "memory");
  // Wait for this wave's async transfer, then cross-wave barrier.
  asm volatile("s_wait_asynccnt 0" ::: "memory");
  __syncthreads();

  // ---- Compute 4 consecutive outputs j = g*1024 + 4t + {0,1,2,3}. ----
  const float k = INV_SQRT2_F;

  const unsigned int b = t >> 3;                  // (4t)>>5 : block-of-32 index
  const float a  = lds[b];
  const float e5 = lds[32u  + b];
  const float e4 = lds[64u  + (t >> 2)];          // (4t)>>4
  const float e3 = lds[128u + (t >> 1)];          // (4t)>>3
  const float e2 = lds[256u + t];                 // (4t)>>2
  const float2 e1 = *(const float2*)&lds[512u + 2u * t];  // (4t)>>1, +1

  // Signs for bits that are constant across the 4 outputs of this thread:
  const float e5s = (t & 4u) ? -e5 : e5;          // bit4(j) = (t>>2)&1
  const float e4s = (t & 2u) ? -e4 : e4;          // bit3(j) = (t>>1)&1
  const float e3s = (t & 1u) ? -e3 : e3;          // bit2(j) =  t&1

  float v = (a + e5s) * k;      // level 5 combine
  v = (v + e4s) * k;            // level 4
  v = (v + e3s) * k;            // level 3
  const float vp = (v + e2) * k;   // level 2, bit1(j)=0  (j=4t,4t+1)
  const float vm = (v - e2) * k;   // level 2, bit1(j)=1  (j=4t+2,4t+3)

  float4 o;
  o.x = (vp + e1.x) * k;        // j even
  o.y = (vp - e1.x) * k;        // j odd
  o.z = (vm + e1.y) * k;
  o.w = (vm - e1.y) * k;

  *(float4*)(out + (unsigned long long)g * 1024ull +
             (unsigned long long)t * 4ull) = o;
}

extern "C" void kernel_launch(void* const* d_in, const int* in_sizes, int n_in,
                              void* d_out, int out_size, void* d_ws, size_t ws_size,
                              hipStream_t stream) {
  const float* x = (const float*)d_in[0];
  float* out = (float*)d_out;
  const unsigned int L = (unsigned int)out_size;   // 33554432 = 2^25
  const unsigned int blocks = L / 1024u;           // 1024 outputs per WG
  ihaar5_fused_kernel<<<blocks, 256, 0, stream>>>(x, out, L);
}